// KNearestNeighbors_20340965114160
// MI455X (gfx1250) — compile-verified
//
#include <hip/hip_runtime.h>

// ---------------------------------------------------------------------------
// Fused KNN (K=16) + EdgeConv feature assembly for B=64, N=2048, C_PTS=3,
// C_FTS=64 on gfx1250 (MI455X).
//
// Strategy (output-bandwidth-bound design):
//  * never materialize the 2048x2048 distance matrix in HBM
//  * distances via V_WMMA_F32_16X16X4_F32 (K padded 3->4 with zeros)
//  * per-row top-17 kept in registers (16 threads/row, strided candidates)
//  * 16-way sorted-list merge per row, then one streaming 1 GiB output pass
//  * output stores are non-temporal (TH=NT) so the 1 GiB stream does not
//    evict the hot 32 MB features array from the 192 MB L2
// ---------------------------------------------------------------------------

typedef __attribute__((ext_vector_type(2))) float v2f;
typedef __attribute__((ext_vector_type(4))) float v4f;   // native vector: OK for
typedef __attribute__((ext_vector_type(8))) float v8f;   // nontemporal builtins

#define NPTS   2048
#define NBATCH 64
#define CFTS   64
#define KNN    16
#define LISTN  (KNN + 1)       // keep self + 16 neighbors
#define CHUNK  256             // candidate columns staged in LDS per pass
#define NCHUNK (NPTS / CHUNK)  // 8
#define DSTRIDE 257            // odd padded row stride (bank-friendly)

__launch_bounds__(256, 2)
__global__ void knn_edge_kernel(const float* __restrict__ pts,
                                const float* __restrict__ fts,
                                float* __restrict__ out) {
    __shared__ float distBuf[16 * DSTRIDE];     // 16 rows x 256 (+pad) dists
    __shared__ float rqLds[16];                 // |q|^2 per query row
    __shared__ float mD[16][16][LISTN];         // per-row per-thread lists
    __shared__ int   mI[16][16][LISTN];
    __shared__ int   nbr[16][KNN];              // final neighbor indices

    const int tid   = threadIdx.x;
    const int lane  = tid & 31;
    const int wave  = tid >> 5;
    const int b     = blockIdx.x >> 7;          // blockIdx.x / 128
    const int qt    = blockIdx.x & 127;
    const int qbase = qt * 16;

    // ---- A operand: 16x4 f32 query tile (ISA layout: VGPR0 = K0|K2, VGPR1 = K1|K3)
    const int    qrow = qbase + (lane & 15);
    const size_t qoff = ((size_t)b * NPTS + qrow) * 3;
    const float qx = pts[qoff + 0];
    const float qy = pts[qoff + 1];
    const float qz = pts[qoff + 2];
    v2f aop;
    aop.x = (lane < 16) ? qx : qz;   // K=0 (x) | K=2 (z)
    aop.y = (lane < 16) ? qy : 0.f;  // K=1 (y) | K=3 (zero pad)

    if (tid < 16) {
        const size_t o = ((size_t)b * NPTS + qbase + tid) * 3;
        const float x = pts[o], y = pts[o + 1], z = pts[o + 2];
        rqLds[tid] = x * x + y * y + z * z;
    }
    __syncthreads();

    // hoist the 8 row-norms this lane's WMMA accumulators correspond to
    const int row0 = (lane < 16) ? 0 : 8;
    float rqv[8];
#pragma unroll
    for (int v = 0; v < 8; ++v) rqv[v] = rqLds[row0 + v];

    // ---- per-thread running top-17 (registers, fully unrolled)
    float kd[LISTN];
    int   ki[LISTN];
#pragma unroll
    for (int i = 0; i < LISTN; ++i) { kd[i] = 3.0e38f; ki[i] = -1; }

    const int selRow = tid >> 4;   // query row this thread selects for
    const int selT   = tid & 15;   // its stride slot within the row

    for (int ch = 0; ch < NCHUNK; ++ch) {
        // ---------- distance tiles via WMMA: wave w owns col-tiles w and w+8
#pragma unroll
        for (int tt = 0; tt < 2; ++tt) {
            const int ct  = wave + tt * 8;
            const int col = ch * CHUNK + ct * 16 + (lane & 15);
            const size_t coff = ((size_t)b * NPTS + col) * 3;
            const float cx = pts[coff + 0];
            const float cy = pts[coff + 1];
            const float cz = pts[coff + 2];
            const float rc = cx * cx + cy * cy + cz * cz;

            v2f bop;                           // B: 4x16 candidate tile
            bop.x = (lane < 16) ? cx : cz;
            bop.y = (lane < 16) ? cy : 0.f;

            v8f acc = {};
            acc = __builtin_amdgcn_wmma_f32_16x16x4_f32(
                /*neg_a=*/false, aop, /*neg_b=*/false, bop,
                /*c_mod=*/(short)0, acc, /*reuse_a=*/false, /*reuse_b=*/false);

            const int cic = ct * 16 + (lane & 15);   // column within chunk
#pragma unroll
            for (int v = 0; v < 8; ++v) {
                // D = |q|^2 + |c|^2 - 2 q.c
                distBuf[(row0 + v) * DSTRIDE + cic] =
                    rqv[v] + rc - 2.0f * acc[v];
            }
        }
        __syncthreads();

        // ---------- top-17 selection: 16 threads per row, stride-16 columns
#pragma unroll
        for (int s = 0; s < 16; ++s) {
            const int   j = selT + s * 16;
            const float d = distBuf[selRow * DSTRIDE + j];
            const int   g = ch * CHUNK + j;
            if (d < kd[LISTN - 1]) {
                kd[LISTN - 1] = d;
                ki[LISTN - 1] = g;
#pragma unroll
                for (int i = LISTN - 1; i > 0; --i) {   // bubble-up, stays sorted
                    if (kd[i] < kd[i - 1]) {
                        float td = kd[i]; kd[i] = kd[i - 1]; kd[i - 1] = td;
                        int   ti = ki[i]; ki[i] = ki[i - 1]; ki[i - 1] = ti;
                    }
                }
            }
        }
        __syncthreads();   // distBuf reused next chunk
    }

    // ---------- publish per-thread sorted lists, then 16-way merge per row
#pragma unroll
    for (int i = 0; i < LISTN; ++i) {
        mD[selRow][selT][i] = kd[i];
        mI[selRow][selT][i] = ki[i];
    }
    __syncthreads();

    if (selT == 0) {
        int head[16];
#pragma unroll
        for (int i = 0; i < 16; ++i) head[i] = 0;
        for (int pick = 0; pick <= KNN; ++pick) {
            float best = 3.0e38f;
            int   bt   = 0;
#pragma unroll
            for (int t2 = 0; t2 < 16; ++t2) {
                const int   h = head[t2];
                const float d = (h < LISTN) ? mD[selRow][t2][h] : 3.0e38f;
                if (d < best) { best = d; bt = t2; }
            }
            const int g = mI[selRow][bt][head[bt]];
#pragma unroll
            for (int t2 = 0; t2 < 16; ++t2) head[t2] += (t2 == bt) ? 1 : 0;
            if (pick > 0) nbr[selRow][pick - 1] = g;   // pick 0 == self, dropped
        }
    }
    __syncthreads();

    // ---------- streaming output: [b][n][k][0:64]=center, [64:128]=nbr-center
    // Non-temporal stores: 1 GiB write-once stream must not rinse the L2
    // (keep the 32 MB, ~17x-reused features array resident instead).
    {
        const int r = tid >> 4;
        const int k = tid & 15;
        const int n = qbase + r;
        const v4f* cf = (const v4f*)(fts + ((size_t)b * NPTS + n) * CFTS);
        const v4f* nf = (const v4f*)(fts + ((size_t)b * NPTS + nbr[r][k]) * CFTS);
        v4f* op = (v4f*)(out + (((size_t)b * NPTS + n) * KNN + k) * (2 * CFTS));
#pragma unroll
        for (int c = 0; c < CFTS / 4; ++c) {
            const v4f cv = cf[c];
            const v4f nv = nf[c];
            const v4f dv = nv - cv;
            __builtin_nontemporal_store(cv, &op[c]);
            __builtin_nontemporal_store(dv, &op[CFTS / 4 + c]);
        }
    }
}

extern "C" void kernel_launch(void* const* d_in, const int* in_sizes, int n_in,
                              void* d_out, int out_size, void* d_ws, size_t ws_size,
                              hipStream_t stream) {
    const float* pts = (const float*)d_in[0];   // [64, 2048, 3]  f32
    const float* fts = (const float*)d_in[1];   // [64, 2048, 64] f32
    float* out = (float*)d_out;                 // [64, 2048, 16, 128] f32
    (void)in_sizes; (void)n_in; (void)out_size; (void)d_ws; (void)ws_size;

    const dim3 grid(NBATCH * (NPTS / 16));      // 8192 workgroups
    const dim3 block(256);                      // 8 waves: one 16-row query tile
    knn_edge_kernel<<<grid, block, 0, stream>>>(pts, fts, out);
}